// FusionGATConv_27212912787629
// MI455X (gfx1250) — compile-verified
//
#include <hip/hip_runtime.h>
#include <hip/hip_bf16.h>

// CDNA5 (gfx1250) fused GATConv-style kernel set.
// Precision: fp32 end-to-end (problem is HBM/atomic bound; 23.3 TB/s * ~1GB
// edge traffic dominates; GEMM uses native V_WMMA_F32_16X16X4_F32).

typedef __attribute__((ext_vector_type(2))) float v2f;
typedef __attribute__((ext_vector_type(8))) float v8f;

#define GAT_INDIM 128
#define GAT_C 64

// ---------------------------------------------------------------------------
// Kernel 1: xp = x @ W  via V_WMMA_F32_16X16X4_F32.
// Block = 256 threads = 8 waves. Each wave computes one 16x16 tile of xp:
//   wave w: rowTile = blockIdx.x*2 + w/4, colTile = w%4  (C=64 -> 4 col tiles)
// W (128x64 f32 = 32KB) staged in LDS once per block.
//
// A-matrix 16x4 f32 layout (ISA 7.12.2): lanes 0-15 rows M=0..15 hold K=0,1
// in VGPR0/1; lanes 16-31 hold K=2,3. -> per lane: contiguous float2 of the
// x row at k + 2*(lane>>4).
// B-matrix 4x16: row K striped across lanes within a VGPR (VGPR0: K=0 lanes
// 0-15 / K=2 lanes 16-31; VGPR1: K=1 / K=3), N = lane&15.
// C/D 16x16: VGPR i holds M=i (lanes 0-15) and M=i+8 (lanes 16-31), N=lane&15.
// ---------------------------------------------------------------------------
__global__ __launch_bounds__(256) void gemm_xw_wmma(
    const float* __restrict__ x, const float* __restrict__ W,
    float* __restrict__ xp, int N) {
  __shared__ float ldsW[GAT_INDIM * GAT_C];  // 32 KB of the 320 KB WGP LDS

  // Cooperative W load: 8192 floats = 2048 float4, 8 per thread.
  {
    const float4* __restrict__ Wv = (const float4*)W;
    float4* Lv = (float4*)ldsW;
#pragma unroll
    for (int i = 0; i < 8; ++i) {
      Lv[threadIdx.x + 256 * i] = Wv[threadIdx.x + 256 * i];
    }
  }
  __syncthreads();

  const int wave = threadIdx.x >> 5;
  const int lane = threadIdx.x & 31;
  const int rowBase = (blockIdx.x * 2 + (wave >> 2)) * 16;
  const int colBase = (wave & 3) * 16;
  if (rowBase >= N) return;  // uniform across the wave (N % 16 == 0)

  const int halfSel = (lane >> 4);      // 0: lanes 0-15, 1: lanes 16-31
  const int kSel = halfSel * 2;         // K offset within the 16x4 A frag
  const int col = colBase + (lane & 15);
  const int row = rowBase + (lane & 15);

  const float* __restrict__ xrow = x + (size_t)row * GAT_INDIM;
  __builtin_prefetch(xrow, 0, 1);  // global_prefetch_b8: pull the 512B row

  v8f acc = {};
#pragma unroll
  for (int k = 0; k < GAT_INDIM; k += 4) {
    v2f a;
    a.x = xrow[k + kSel];
    a.y = xrow[k + kSel + 1];
    v2f b;
    b.x = ldsW[(k + kSel) * GAT_C + col];
    b.y = ldsW[(k + kSel + 1) * GAT_C + col];
    acc = __builtin_amdgcn_wmma_f32_16x16x4_f32(
        /*neg_a=*/false, a, /*neg_b=*/false, b,
        /*c_mod=*/(short)0, acc, /*reuse_a=*/false, /*reuse_b=*/false);
  }

#pragma unroll
  for (int i = 0; i < 8; ++i) {
    const int r = rowBase + i + 8 * halfSel;
    xp[(size_t)r * GAT_C + col] = acc[i];
  }
}

// ---------------------------------------------------------------------------
// Kernel 2: per-node attention scores a_src[n] = dot(xp[n,:], att_src),
//           a_dst[n] = dot(xp[n,:], att_dst).  att vectors cached in LDS.
// ---------------------------------------------------------------------------
__global__ __launch_bounds__(256) void node_scores(
    const float* __restrict__ xp, const float* __restrict__ att_src,
    const float* __restrict__ att_dst, float* __restrict__ a_src,
    float* __restrict__ a_dst, int N) {
  __shared__ float s_as[GAT_C];
  __shared__ float s_ad[GAT_C];
  if (threadIdx.x < GAT_C) {
    s_as[threadIdx.x] = att_src[threadIdx.x];
    s_ad[threadIdx.x] = att_dst[threadIdx.x];
  }
  __syncthreads();

  const int n = blockIdx.x * 256 + threadIdx.x;
  if (n >= N) return;
  const float4* __restrict__ row = (const float4*)(xp + (size_t)n * GAT_C);
  const float4* sa = (const float4*)s_as;
  const float4* sd = (const float4*)s_ad;
  float ss = 0.f, dd = 0.f;
#pragma unroll
  for (int i = 0; i < GAT_C / 4; ++i) {
    float4 v = row[i];
    float4 a = sa[i];
    float4 b = sd[i];
    ss += v.x * a.x + v.y * a.y + v.z * a.z + v.w * a.w;
    dd += v.x * b.x + v.y * b.y + v.z * b.z + v.w * b.w;
  }
  a_src[n] = ss;
  a_dst[n] = dd;
}

// ---------------------------------------------------------------------------
// Kernel 3: re-init accumulators every launch (atomics below accumulate).
// seg-max stored as int bits; sigmoid output is in (0,1) so positive-float
// ordering == signed-int ordering; init bits to 0 (== 0.0f).
// ---------------------------------------------------------------------------
__global__ __launch_bounds__(256) void init_buffers(
    float* __restrict__ out, int* __restrict__ m_bits,
    float* __restrict__ denom, int N) {
  const long long idx = (long long)blockIdx.x * 256 + threadIdx.x;
  const long long total = (long long)N * GAT_C;
  if (idx < total) out[idx] = 0.f;
  if (idx < N) {
    m_bits[idx] = 0;
    denom[idx] = 0.f;
  }
}

// ---------------------------------------------------------------------------
// Kernel 4: e = sigmoid(a_src[src] + a_dst[dst]); segment max over dst.
// ---------------------------------------------------------------------------
__global__ __launch_bounds__(256) void edge_pass1(
    const int* __restrict__ ei, const float* __restrict__ a_src,
    const float* __restrict__ a_dst, float* __restrict__ e_buf,
    int* __restrict__ m_bits, int E) {
  const int t = blockIdx.x * 256 + threadIdx.x;
  if (t >= E) return;
  const int s = ei[t];
  const int d = ei[E + t];
  const float z = a_src[s] + a_dst[d];
  const float e = 1.f / (1.f + __expf(-z));
  e_buf[t] = e;
  atomicMax(&m_bits[d], __float_as_int(e));
}

// ---------------------------------------------------------------------------
// Kernel 5: ex = exp(e - m[dst]); segment sum of ex into denom.
// ---------------------------------------------------------------------------
__global__ __launch_bounds__(256) void edge_pass2(
    const int* __restrict__ ei, const float* __restrict__ e_buf,
    const float* __restrict__ m, float* __restrict__ ex_buf,
    float* __restrict__ denom, int E) {
  const int t = blockIdx.x * 256 + threadIdx.x;
  if (t >= E) return;
  const int d = ei[E + t];
  const float ex = __expf(e_buf[t] - m[d]);
  ex_buf[t] = ex;
  atomicAdd(&denom[d], ex);
}

// ---------------------------------------------------------------------------
// Kernel 6: gamma = lam*alpha + (1-lam)*beta; out[dst] += xp[src]*gamma.
// One wave32 per edge, 2 channels per lane -> 64 f32 atomics per edge.
// ---------------------------------------------------------------------------
__global__ __launch_bounds__(256) void edge_scatter(
    const int* __restrict__ ei, const float* __restrict__ ex_buf,
    const float* __restrict__ denom, const float* __restrict__ beta,
    const float* __restrict__ lam01, const float* __restrict__ xp,
    float* __restrict__ out, int E) {
  const int edge = blockIdx.x * 8 + (threadIdx.x >> 5);
  const int lane = threadIdx.x & 31;
  if (edge >= E) return;
  const int s = ei[edge];
  const int d = ei[E + edge];
  const float lam = lam01[0];
  const float alpha = ex_buf[edge] / (denom[d] + 1e-16f);
  const float gamma = lam * alpha + (1.f - lam) * beta[edge];
  const float* __restrict__ xr = xp + (size_t)s * GAT_C;
  float* orow = out + (size_t)d * GAT_C;
  atomicAdd(&orow[lane], xr[lane] * gamma);
  atomicAdd(&orow[lane + 32], xr[lane + 32] * gamma);
}

// ---------------------------------------------------------------------------
// Launch. Inputs (setup_inputs order):
//   0: x        [N*128]  f32
//   1: edge_idx [2*E]    int
//   2: beta     [E]      f32
//   3: lam01    [1]      f32
//   4: W        [128*64] f32
//   5: att_src  [64]     f32
//   6: att_dst  [64]     f32
// Output: [N*64] f32. Workspace layout (floats): xp | a_src | a_dst | m |
// denom | e_buf | ex_buf  (~40 MB).
// ---------------------------------------------------------------------------
extern "C" void kernel_launch(void* const* d_in, const int* in_sizes, int n_in,
                              void* d_out, int out_size, void* d_ws, size_t ws_size,
                              hipStream_t stream) {
  const float* x = (const float*)d_in[0];
  const int* ei = (const int*)d_in[1];
  const float* beta = (const float*)d_in[2];
  const float* lam01 = (const float*)d_in[3];
  const float* W = (const float*)d_in[4];
  const float* att_src = (const float*)d_in[5];
  const float* att_dst = (const float*)d_in[6];
  float* out = (float*)d_out;

  const int N = in_sizes[0] / GAT_INDIM;
  const int E = in_sizes[2];

  float* ws = (float*)d_ws;
  float* xp = ws;                       ws += (size_t)N * GAT_C;
  float* a_src = ws;                    ws += N;
  float* a_dst = ws;                    ws += N;
  float* m = ws;                        ws += N;   // int-bit seg-max
  float* denom = ws;                    ws += N;
  float* e_buf = ws;                    ws += E;
  float* ex_buf = ws;                   ws += E;

  // 1) xp = x @ W (WMMA f32). Each block covers 32 rows (2 tiles x 8 waves/4).
  const int gemmBlocks = (N + 31) / 32;
  gemm_xw_wmma<<<gemmBlocks, 256, 0, stream>>>(x, W, xp, N);

  // 2) per-node scores
  node_scores<<<(N + 255) / 256, 256, 0, stream>>>(xp, att_src, att_dst,
                                                   a_src, a_dst, N);

  // 3) zero accumulators (must happen every call: graph replays accumulate)
  const long long total = (long long)N * GAT_C;
  init_buffers<<<(unsigned)((total + 255) / 256), 256, 0, stream>>>(
      out, (int*)m, denom, N);

  // 4) sigmoid + segment max
  edge_pass1<<<(E + 255) / 256, 256, 0, stream>>>(ei, a_src, a_dst, e_buf,
                                                  (int*)m, E);

  // 5) exp + segment sum
  edge_pass2<<<(E + 255) / 256, 256, 0, stream>>>(ei, e_buf, m, ex_buf,
                                                  denom, E);

  // 6) normalize, blend with beta, scatter-add messages
  edge_scatter<<<(E + 7) / 8, 256, 0, stream>>>(ei, ex_buf, denom, beta,
                                                lam01, xp, out, E);
}